// FitnessHead_40785009443223
// MI455X (gfx1250) — compile-verified
//
#include <hip/hip_runtime.h>
#include <cstdint>

#define C22    22
#define CCSQ   484                      // 22*22 floats per (i,j) block
#define NRES   384
#define NVAR   64
#define JT     8                        // j-blocks staged per tile
#define JSPLIT 2                        // j-range split across blocks
#define JHALF  (NRES / JSPLIT)          // 192
#define NTILE  (JHALF / JT)             // 24 tiles per block
#define CHUNKS ((JT * CCSQ * 4) / 16)   // 968 x 16B async chunks per tile
#define MN     (NVAR * NRES)            // 24576

typedef __attribute__((ext_vector_type(2))) float v2f;
typedef __attribute__((ext_vector_type(8))) float v8f;

// ---------------------------------------------------------------------------
// Stage one 8-block tile (15488 B) into LDS with async global->LDS loads.
// Every wave issues EXACTLY 4 async instructions (out-of-range lanes redo
// chunk 0 with identical bytes) so ASYNCcnt arithmetic is exact: with two
// tiles in flight, s_wait_asynccnt 4 == "all but the newest tile landed"
// (async loads complete in order). NT hint: 285MB streamed once must not
// thrash the 192MB L2 (variant/logits stay resident).
// ---------------------------------------------------------------------------
__device__ __forceinline__ void async_stage_tile(const float* gsrc,
                                                 unsigned lds_base,
                                                 int tid) {
    const unsigned long long saddr = (unsigned long long)(uintptr_t)gsrc;
    #pragma unroll
    for (int kk = 0; kk < 4; ++kk) {
        const int k = tid + kk * 256;
        const unsigned voff = (k < CHUNKS) ? (unsigned)k * 16u : 0u;
        const unsigned ldsa = lds_base + voff;
        asm volatile("global_load_async_to_lds_b128 %0, %1, %2 th:TH_LOAD_NT"
                     :: "v"(ldsa), "v"(voff), "s"(saddr) : "memory");
    }
}

// ---------------------------------------------------------------------------
// Kernel 1: bandwidth-bound double gather, depth-2 async pipeline.
// Block (i, half) accumulates sum_{j in half} eij5[i,j,v[m,i],v[m,j]] for all
// 64 variants; partial sums -> d_ws[half][m][i]. 768 blocks x ~31KB in
// flight ~= 24MB outstanding -> enough to saturate 23.3 TB/s.
// ---------------------------------------------------------------------------
__global__ __launch_bounds__(256)
void fitness_gather_kernel(const float* __restrict__ eij,
                           const int*   __restrict__ variant,
                           float*       __restrict__ partial)
{
    __shared__ __align__(16) float buf[3][JT * CCSQ];   // 3 x 15488 B
    __shared__ float red[256];

    const int half  = blockIdx.x & 1;
    const int i     = blockIdx.x >> 1;     // residue index
    const int jbase = half * JHALF;
    const int t     = threadIdx.x;
    const int m     = t & 63;              // variant handled by this thread
    const int slot  = t >> 6;              // 0..3: j sub-slot

    const int cbase = variant[m * NRES + i] * C22;
    const float* esrc = eij + ((size_t)i * NRES + jbase) * CCSQ;

    // prime the pipeline: tiles 0 and 1 in flight
    async_stage_tile(esrc,             (unsigned)(uintptr_t)&buf[0][0], t);
    async_stage_tile(esrc + JT * CCSQ, (unsigned)(uintptr_t)&buf[1][0], t);

    float acc = 0.0f;
    for (int tile = 0; tile < NTILE; ++tile) {
        const int bb = tile % 3;
        if (tile < NTILE - 1) {
            // newest tile (tile+1, 4 instrs/wave) may stay in flight
            asm volatile("s_wait_asynccnt 0x4" ::: "memory");
        } else {
            asm volatile("s_wait_asynccnt 0x0" ::: "memory");
        }
        __syncthreads();                   // everyone's tile data landed
        if (tile + 2 < NTILE) {
            async_stage_tile(esrc + (size_t)(tile + 2) * JT * CCSQ,
                             (unsigned)(uintptr_t)&buf[(tile + 2) % 3][0], t);
        }
        const int j0 = jbase + tile * JT;
        #pragma unroll
        for (int jj2 = 0; jj2 < 2; ++jj2) {
            const int jj = slot * 2 + jj2;
            const int d  = variant[m * NRES + j0 + jj];
            acc += buf[bb][jj * CCSQ + cbase + d];
        }
        __syncthreads();   // all reads of buf[bb] done before it is refilled
    }

    red[t] = acc;
    __syncthreads();
    if (t < NVAR) {
        partial[half * MN + t * NRES + i] =
            red[t] + red[t + 64] + red[t + 128] + red[t + 192];
    }
}

// ---------------------------------------------------------------------------
// Kernel 2: combine j-halves, add per-residue bias, apply sigma.
// ---------------------------------------------------------------------------
__global__ __launch_bounds__(256)
void fitness_epilogue_kernel(const float* __restrict__ partial,
                             const float* __restrict__ ei,
                             const int*   __restrict__ variant,
                             const float* __restrict__ sigma_w,
                             const float* __restrict__ sigma_b,
                             float*       __restrict__ logits)
{
    const int idx = blockIdx.x * 256 + threadIdx.x;   // m*NRES + i
    if (idx >= MN) return;
    const int i  = idx % NRES;
    const int ci = variant[idx];
    const float s = partial[idx] + partial[MN + idx];
    logits[idx] = (ei[i * C22 + ci] + s) * sigma_w[0] + sigma_b[0];
}

// ---------------------------------------------------------------------------
// Kernel 3: variant_logit[m] = sum_i (logits[m,i]-logits[0,i])*mask[m,i]
// as a real (tiny) matmul: A = delta*mask [16m x 4i tiles], B = ones(4x16),
// accumulated with V_WMMA_F32_16X16X4_F32; take D column N=0.
// ---------------------------------------------------------------------------
__global__ __launch_bounds__(128)
void fitness_reduce_wmma_kernel(const float* __restrict__ logits,
                                const float* __restrict__ mask,
                                float*       __restrict__ vout)
{
    const int lane  = threadIdx.x & 31;
    const int wv    = threadIdx.x >> 5;     // m-tile 0..3 (one per wave32)
    const int Mrow  = lane & 15;            // A row M = lane%16
    const int khalf = lane >> 4;            // lanes 16-31 hold K=2,3
    const int m     = wv * 16 + Mrow;

    v8f c = {0.f, 0.f, 0.f, 0.f, 0.f, 0.f, 0.f, 0.f};
    v2f bones; bones.x = 1.0f; bones.y = 1.0f;   // B layout irrelevant: all ones

    for (int i0 = 0; i0 < NRES; i0 += 4) {
        const int ia = i0 + khalf * 2;
        v2f a;
        a.x = (logits[m * NRES + ia]     - logits[ia])     * mask[m * NRES + ia];
        a.y = (logits[m * NRES + ia + 1] - logits[ia + 1]) * mask[m * NRES + ia + 1];
        // 8-arg form: (neg_a, A, neg_b, B, c_mod, C, reuse_a, reuse_b)
        c = __builtin_amdgcn_wmma_f32_16x16x4_f32(false, a, false, bones,
                                                  (short)0, c, false, false);
    }

    // D 16x16 f32: lanes 0-15 -> N=lane, VGPR v -> M=v; lanes 16-31 -> M=v+8.
    // Column N=0 lives in lanes 0 (rows 0-7) and 16 (rows 8-15).
    if ((lane & 15) == 0) {
        const int rbase = wv * 16 + khalf * 8;
        #pragma unroll
        for (int v = 0; v < 8; ++v) vout[rbase + v] = c[v];
    }
}

// ---------------------------------------------------------------------------
extern "C" void kernel_launch(void* const* d_in, const int* in_sizes, int n_in,
                              void* d_out, int out_size, void* d_ws, size_t ws_size,
                              hipStream_t stream) {
    const float* eij     = (const float*)d_in[0];   // [1,384,384,484] f32
    const float* ei      = (const float*)d_in[1];   // [1,384,22]      f32
    const int*   variant = (const int*)  d_in[2];   // [1,64,384]      i32
    const float* vmask   = (const float*)d_in[3];   // [1,64,384]      f32
    const float* sw      = (const float*)d_in[4];   // [1,1]           f32
    const float* sb      = (const float*)d_in[5];   // [1]             f32

    float* out     = (float*)d_out;   // [64*384] logits ++ [64] variant_logit
    float* partial = (float*)d_ws;    // [2][64][384] f32 = 192KB scratch

    fitness_gather_kernel<<<NRES * JSPLIT, 256, 0, stream>>>(eij, variant, partial);
    fitness_epilogue_kernel<<<(MN + 255) / 256, 256, 0, stream>>>(
        partial, ei, variant, sw, sb, out);
    fitness_reduce_wmma_kernel<<<1, 128, 0, stream>>>(out, vmask, out + MN);
}